// Attention_4681514353180
// MI455X (gfx1250) — compile-verified
//
#include <hip/hip_runtime.h>
#include <math.h>

#define N_ 64
#define T_ 4096
#define D_ 128
#define ND_ (N_ * D_)      // 8192 floats: stride between consecutive t rows
#define S_ 16              // t-chunks for the value pass
#define CHUNK_ (T_ / S_)   // 256 t per chunk

typedef float v2f __attribute__((ext_vector_type(2)));
typedef float v8f __attribute__((ext_vector_type(8)));

// ---------------------------------------------------------------------------
// Kernel 1: energy[n,t] = <key[t,n,:], query[n,:]>  via V_WMMA_F32_16X16X4_F32
// A = query row-replicated (16x4), B = 16 key rows (4x16), 32 K-steps over D.
// grid = (N, T/512), block = 256 (8 waves); each wave owns 4 tiles of 16 t's.
// ---------------------------------------------------------------------------
__global__ __launch_bounds__(256) void k_energy(const float* __restrict__ query,
                                                const float* __restrict__ key,
                                                float* __restrict__ energy) {
    __shared__ float qs[D_];
    const int n   = blockIdx.x;
    const int tid = threadIdx.x;
    if (tid < D_) qs[tid] = query[n * D_ + tid];
    __syncthreads();

    const int w = tid >> 5, lane = tid & 31, lo = lane & 15, hi = lane >> 4;

    for (int i = 0; i < 4; ++i) {
        const int t0 = blockIdx.y * 512 + (w + 8 * i) * 16;
        v8f c = {0.f, 0.f, 0.f, 0.f, 0.f, 0.f, 0.f, 0.f};
        // lane reads key[t0+lo][n][d + 2*hi .. +1] : contiguous float2 per step
        const float* kb = key + (size_t)(t0 + lo) * ND_ + (size_t)n * D_ + 2 * hi;
#pragma unroll 8
        for (int d = 0; d < D_; d += 4) {
            v2f a, b;
            a[0] = qs[d + 2 * hi];          // A[m, K=2*hi]   (row-replicated q)
            a[1] = qs[d + 1 + 2 * hi];      // A[m, K=2*hi+1]
            const float2 kk = *(const float2*)(kb + d);
            b[0] = kk.x;                    // B[K=2*hi,   N=lo]
            b[1] = kk.y;                    // B[K=2*hi+1, N=lo]
            c = __builtin_amdgcn_wmma_f32_16x16x4_f32(false, a, false, b,
                                                      (short)0, c, false, false);
        }
        // every C row equals the energy row; C vgpr0, lanes 0-15 = (M=0, N=lane)
        if (lane < 16) energy[(size_t)n * T_ + t0 + lane] = c[0];
    }
}

// ---------------------------------------------------------------------------
// Kernel 2: mask output + softmax stats per row n.
// me[t] = mask*energy (multiplicative mask, NOT -inf — faithful to reference).
// s1 = sum exp(me-max);  s2 = sum (exp(me-max)/s1)  (the redundant renorm);
// rowinv = 1/(s1*s2).  grid = N blocks of 256.
// ---------------------------------------------------------------------------
__global__ __launch_bounds__(256) void k_stats(const float* __restrict__ energy,
                                               const long long* __restrict__ slen,
                                               float* __restrict__ mask_out,
                                               float* __restrict__ rowmax,
                                               float* __restrict__ rowinv) {
    __shared__ float sdata[256];
    const int n   = blockIdx.x;
    const int tid = threadIdx.x;
    const int len = (int)slen[n];

    float m = -INFINITY;
    for (int t = tid; t < T_; t += 256) {
        const float e  = energy[(size_t)n * T_ + t];
        const float me = (t < len) ? e : 0.0f;
        m = fmaxf(m, me);
        mask_out[(size_t)n * T_ + t] = (t < len) ? 1.0f : 0.0f;
    }
    sdata[tid] = m; __syncthreads();
    for (int s = 128; s > 0; s >>= 1) {
        if (tid < s) sdata[tid] = fmaxf(sdata[tid], sdata[tid + s]);
        __syncthreads();
    }
    m = sdata[0]; __syncthreads();

    float s1 = 0.f;
    for (int t = tid; t < T_; t += 256) {
        const float e  = energy[(size_t)n * T_ + t];
        const float me = (t < len) ? e : 0.0f;
        s1 += expf(me - m);
    }
    sdata[tid] = s1; __syncthreads();
    for (int s = 128; s > 0; s >>= 1) {
        if (tid < s) sdata[tid] += sdata[tid + s];
        __syncthreads();
    }
    s1 = sdata[0]; __syncthreads();

    float s2 = 0.f;
    for (int t = tid; t < T_; t += 256) {
        const float e  = energy[(size_t)n * T_ + t];
        const float me = (t < len) ? e : 0.0f;
        s2 += expf(me - m) / s1;
    }
    sdata[tid] = s2; __syncthreads();
    for (int s = 128; s > 0; s >>= 1) {
        if (tid < s) sdata[tid] += sdata[tid + s];
        __syncthreads();
    }
    if (tid == 0) {
        rowmax[n] = m;
        rowinv[n] = 1.0f / (s1 * sdata[0]);
    }
}

// ---------------------------------------------------------------------------
// Kernel 3: partial context: sum over 256 t's of exp(me-max) * value[t,n,:]
// via WMMA: A = softmax-weight row-replicated (16x4), B = 4x16 value tile.
// grid = (N, S_), block = 256 (8 waves; wave owns 32 t's, 8 column tiles).
// ---------------------------------------------------------------------------
__global__ __launch_bounds__(256) void k_context(const float* __restrict__ value,
                                                 const float* __restrict__ energy,
                                                 const long long* __restrict__ slen,
                                                 const float* __restrict__ rowmax,
                                                 float* __restrict__ partial) {
    __shared__ float ps[CHUNK_];
    __shared__ float red[8][D_];
    const int n   = blockIdx.x;
    const int s   = blockIdx.y;
    const int tid = threadIdx.x;
    const int len = (int)slen[n];
    const float mx = rowmax[n];

    {   // un-normalized softmax weights for this t-chunk
        const int t    = s * CHUNK_ + tid;
        const float e  = energy[(size_t)n * T_ + t];
        const float me = (t < len) ? e : 0.0f;
        ps[tid] = expf(me - mx);
    }
    __syncthreads();

    const int w = tid >> 5, lane = tid & 31, lo = lane & 15, hi = lane >> 4;
    const int tw = w * 32;  // this wave's local t base within the chunk

    const v8f zero = {0.f, 0.f, 0.f, 0.f, 0.f, 0.f, 0.f, 0.f};
    v8f acc[8];
#pragma unroll
    for (int ct = 0; ct < 8; ++ct) acc[ct] = zero;

    const size_t vbase = (size_t)(s * CHUNK_ + tw) * ND_ + (size_t)n * D_;
    for (int kt = 0; kt < 32; kt += 4) {
        v2f a;
        a[0] = ps[tw + kt + 2 * hi];        // A[m, K] = weight, row-replicated
        a[1] = ps[tw + kt + 1 + 2 * hi];
        const float* v0 = value + vbase + (size_t)(kt + 2 * hi) * ND_ + lo;
        const float* v1 = v0 + ND_;
#pragma unroll
        for (int ct = 0; ct < 8; ++ct) {    // 8 tiles of 16 output features
            v2f b;
            b[0] = v0[ct * 16];             // B[K=2*hi,   N=lo] = value row t
            b[1] = v1[ct * 16];             // B[K=2*hi+1, N=lo] = value row t+1
            acc[ct] = __builtin_amdgcn_wmma_f32_16x16x4_f32(false, a, false, b,
                                                            (short)0, acc[ct],
                                                            false, false);
        }
    }
    // every C row is identical; C vgpr0 lanes 0-15 = features d0..d0+15
    if (lane < 16) {
#pragma unroll
        for (int ct = 0; ct < 8; ++ct) red[w][ct * 16 + lo] = acc[ct][0];
    }
    __syncthreads();
    if (tid < D_) {
        float sum = 0.f;
#pragma unroll
        for (int w2 = 0; w2 < 8; ++w2) sum += red[w2][tid];
        partial[((size_t)s * N_ + n) * D_ + tid] = sum;
    }
}

// ---------------------------------------------------------------------------
// Kernel 4: reduce the S_ partials and apply the softmax normalization.
// ---------------------------------------------------------------------------
__global__ __launch_bounds__(128) void k_reduce(const float* __restrict__ partial,
                                                const float* __restrict__ rowinv,
                                                float* __restrict__ ctx_out) {
    const int n = blockIdx.x, d = threadIdx.x;
    float sum = 0.f;
#pragma unroll
    for (int s = 0; s < S_; ++s) sum += partial[((size_t)s * N_ + n) * D_ + d];
    ctx_out[(size_t)n * D_ + d] = sum * rowinv[n];
}

extern "C" void kernel_launch(void* const* d_in, const int* in_sizes, int n_in,
                              void* d_out, int out_size, void* d_ws, size_t ws_size,
                              hipStream_t stream) {
    const float*     query = (const float*)d_in[0];          // (N, D)
    const float*     key   = (const float*)d_in[1];          // (T, N, D)
    const float*     value = (const float*)d_in[2];          // (T, N, D)
    const long long* slen  = (const long long*)d_in[3];      // (N,) int64

    float* out      = (float*)d_out;
    float* ctx_out  = out;                 // (N, D)  first output
    float* mask_out = out + N_ * D_;       // (N, T)  second output

    float* ws      = (float*)d_ws;
    float* energy  = ws;                   // N*T   floats (1 MB, L2-resident)
    float* rowmax  = ws + (size_t)N_ * T_; // N
    float* rowinv  = rowmax + N_;          // N
    float* partial = rowinv + N_;          // S_*N*D floats (512 KB)

    k_energy <<<dim3(N_, T_ / 512), 256, 0, stream>>>(query, key, energy);
    k_stats  <<<dim3(N_),           256, 0, stream>>>(energy, slen, mask_out,
                                                      rowmax, rowinv);
    k_context<<<dim3(N_, S_),       256, 0, stream>>>(value, energy, slen,
                                                      rowmax, partial);
    k_reduce <<<dim3(N_),           128, 0, stream>>>(partial, rowinv, ctx_out);
}